// PerlinPowerFractal_223338299477
// MI455X (gfx1250) — compile-verified
//
#include <hip/hip_runtime.h>
#include <stdint.h>

#define HH 1024
#define WW 1024
#define BATCH 16
#define OCTAVES 8
#define PCOUNT (1024 * 1024)

typedef unsigned int v4u __attribute__((ext_vector_type(4)));
typedef int v8i __attribute__((ext_vector_type(8)));
typedef int v4i __attribute__((ext_vector_type(4)));

// Order-preserving float<->uint encoding so uint atomicMin/Max give float min/max.
__device__ __forceinline__ unsigned fenc(float f) {
    unsigned u = __float_as_uint(f);
    return (u & 0x80000000u) ? ~u : (u | 0x80000000u);
}
__device__ __forceinline__ float fdec(unsigned u) {
    u = (u & 0x80000000u) ? (u & 0x7fffffffu) : ~u;
    return __uint_as_float(u);
}

__device__ __forceinline__ float fade_(float t) {
    return t * t * t * (t * (t * 6.0f - 15.0f) + 10.0f);
}
__device__ __forceinline__ float lerp_(float t, float a, float b) {
    return a + t * (b - a);
}
__device__ __forceinline__ float grad_(int h, float x, float y, float z) {
    h &= 15;
    float u = (h < 8) ? x : y;
    float v = (h < 4) ? y : (((h == 12) | (h == 14)) ? x : z);
    float r = (h & 1) ? -u : u;
    return r + ((h & 2) ? -v : v);
}

__global__ void perlin_init(unsigned* __restrict__ mmx) {
    int t = threadIdx.x;
    if (t < 2 * BATCH) mmx[t] = (t & 1) ? 0u : 0xffffffffu;  // [2b]=min slot, [2b+1]=max slot
}

__global__ __launch_bounds__(256) void perlin_main(
    const int* __restrict__ p_all, const float* __restrict__ Xp,
    const float* __restrict__ Yp, const float* __restrict__ Zp,
    float* __restrict__ out, unsigned* __restrict__ mmx) {
    __shared__ int lds_p[256];
    __shared__ float smin[8], smax[8];

    const int b = blockIdx.y;
    const int* __restrict__ pb = p_all + (long long)b * (2 * PCOUNT);

    // --- Stage p[0..255] (first-level stripe, indices bounded by 255) into LDS ---
    // CDNA5 path: TDM tensor_load_to_lds of a 256x1 4-byte tile, issued by wave 0.
    if (threadIdx.x < 32) {
        unsigned ldsAddr = (unsigned)(uintptr_t)(&lds_p[0]);  // flat LDS addr low 32 bits = LDS offset
        unsigned long long ga = (unsigned long long)(uintptr_t)pb;
        v4u g0;
        v8i g1;
        v4i g2 = {0, 0, 0, 0};
        v4i g3 = {0, 0, 0, 0};
        v8i g4 = {0, 0, 0, 0, 0, 0, 0, 0};
        g0[0] = 1u;                                                // count=1 (valid), user mode
        g0[1] = ldsAddr;                                           // lds_addr (bytes)
        g0[2] = (unsigned)(ga & 0xffffffffu);                      // global_addr[31:0]
        g0[3] = (unsigned)((ga >> 32) & 0x1ffffffu) | (2u << 30);  // global_addr[56:32] | type=2
        g1[0] = (2 << 16);                                         // data_size=4B, workgroup_mask=0
        g1[1] = (int)(256u << 16);                                 // tensor_dim0[15:0]=256
        g1[2] = (int)(1u << 16);                                   // tensor_dim0 hi=0, tensor_dim1[15:0]=1
        g1[3] = (int)(256u << 16);                                 // tensor_dim1 hi=0, tile_dim0=256
        g1[4] = 1;                                                 // tile_dim1=1, tile_dim2=0
        g1[5] = 256;                                               // tensor_dim0_stride[31:0]=256
        g1[6] = (int)(256u << 16);                                 // stride hi=0, tensor_dim1_stride lo=256
        g1[7] = 0;
        __builtin_amdgcn_tensor_load_to_lds(g0, g1, g2, g3, g4, 0);
        __builtin_amdgcn_s_wait_tensorcnt(0);
    }
    // Idempotent plain fill of the same bytes (correctness does not depend on the TDM op).
    lds_p[threadIdx.x] = pb[threadIdx.x];
    __syncthreads();

    const float Xv = Xp[0], Yv = Yp[0], Zv = Zp[0];
    const int pix = blockIdx.x * 256 + threadIdx.x;
    const int row = pix >> 10;
    const int col = pix & 1023;
    const float xb = ((float)col + Xv) * 0.01f;
    const float yb = ((float)row + Yv) * 0.01f;
    const float zb = (0.1f * (float)b + Zv) * 0.01f;

    float acc = 0.0f;
    float freq = 1.0f;
    float wgt = 1.0f;  // amp^(EXPONENT+1) = 0.5^(5o)
#pragma unroll
    for (int o = 0; o < OCTAVES; ++o) {
        float x = xb * freq, y = yb * freq, z = zb * freq;
        float fx = floorf(x), fy = floorf(y), fz = floorf(z);
        int Xi = ((int)fx) % 255;
        int Yi = ((int)fy) % 255;
        int Zi = ((int)fz) % 255;
        float xf = x - fx, yf = y - fy, zf = z - fz;
        float u = fade_(xf), v = fade_(yf), w = fade_(zf);

        int A = lds_p[Xi] + Yi;        // first level from LDS
        int Bc = lds_p[Xi + 1] + Yi;
        int pA = pb[A];                // second level: 4 L2 gathers
        int pA1 = pb[A + 1];
        int pB = pb[Bc];
        int pB1 = pb[Bc + 1];
        int AA = pA + Zi, AB = pA1 + Zi, BA = pB + Zi, BB = pB1 + Zi;
        // third level: 8 L2 gathers
        float x1 = lerp_(u, grad_(pb[AA], xf, yf, zf), grad_(pb[BA], xf - 1.f, yf, zf));
        float x2 = lerp_(u, grad_(pb[AB], xf, yf - 1.f, zf), grad_(pb[BB], xf - 1.f, yf - 1.f, zf));
        float x3 = lerp_(u, grad_(pb[AA + 1], xf, yf, zf - 1.f), grad_(pb[BA + 1], xf - 1.f, yf, zf - 1.f));
        float x4 = lerp_(u, grad_(pb[AB + 1], xf, yf - 1.f, zf - 1.f), grad_(pb[BB + 1], xf - 1.f, yf - 1.f, zf - 1.f));
        float n = lerp_(w, lerp_(v, x1, x2), lerp_(v, x3, x4));

        acc += n * wgt;
        freq *= 2.0f;
        wgt *= 0.03125f;  // *= 0.5^5
    }

    out[((size_t)b << 20) + pix] = acc;

    // --- Per-image min/max reduction (wave32 shuffles -> LDS -> one atomic per block) ---
    float vmin = acc, vmax = acc;
#pragma unroll
    for (int off = 16; off > 0; off >>= 1) {
        vmin = fminf(vmin, __shfl_down(vmin, off, 32));
        vmax = fmaxf(vmax, __shfl_down(vmax, off, 32));
    }
    int wid = threadIdx.x >> 5, lane = threadIdx.x & 31;
    if (lane == 0) { smin[wid] = vmin; smax[wid] = vmax; }
    __syncthreads();
    if (threadIdx.x == 0) {
#pragma unroll
        for (int i = 1; i < 8; ++i) {
            vmin = fminf(vmin, smin[i]);
            vmax = fmaxf(vmax, smax[i]);
        }
        atomicMin(&mmx[2 * b], fenc(vmin));
        atomicMax(&mmx[2 * b + 1], fenc(vmax));
    }
}

__global__ __launch_bounds__(256) void perlin_norm(float* __restrict__ out,
                                                   const unsigned* __restrict__ mmx) {
    size_t idx = (size_t)blockIdx.x * 256 + threadIdx.x;
    int b = (int)(idx >> 20);
    float mn = fdec(mmx[2 * b]);
    float mx = fdec(mmx[2 * b + 1]);
    float inv = 1.0f / (mx - mn);
    out[idx] = (out[idx] - mn) * inv;  // brightness/contrast/clamp are identity here
}

extern "C" void kernel_launch(void* const* d_in, const int* in_sizes, int n_in,
                              void* d_out, int out_size, void* d_ws, size_t ws_size,
                              hipStream_t stream) {
    const int* p_all = (const int*)d_in[0];
    const float* X = (const float*)d_in[1];
    const float* Y = (const float*)d_in[2];
    const float* Z = (const float*)d_in[3];
    float* out = (float*)d_out;
    unsigned* mmx = (unsigned*)d_ws;

    perlin_init<<<1, 64, 0, stream>>>(mmx);
    perlin_main<<<dim3((HH * WW) / 256, BATCH), 256, 0, stream>>>(p_all, X, Y, Z, out, mmx);
    perlin_norm<<<(unsigned)(((size_t)BATCH * HH * WW) / 256), 256, 0, stream>>>(out, mmx);
}